// GumbelGCN_6476810682534
// MI455X (gfx1250) — compile-verified
//
#include <hip/hip_runtime.h>
#include <cstdint>
#include <cstddef>

// ---- problem constants (match reference) ----
#define NEGV  (-1000000000.0f)
#define KSEL  16
#define DF    8      // node feature dim
#define HF    16     // hidden dims H1 == H2
#define OUTF  112    // output dim
#define ROWB  256    // threads per row-block (4096/256 = 16 cols per thread)

typedef float v2f __attribute__((ext_vector_type(2)));
typedef float v8f __attribute__((ext_vector_type(8)));

// ---------------------------------------------------------------------------
// 1) per-node score precompute: s_node = x @ w_node, s_nbr = x @ w_nbr
// ---------------------------------------------------------------------------
__global__ void k_node_scores(const float* __restrict__ x, const float* __restrict__ mlp_w,
                              float* __restrict__ s_node, float* __restrict__ s_nbr, int N) {
  int i = blockIdx.x * blockDim.x + threadIdx.x;
  if (i >= N) return;
  float a = 0.f, b = 0.f;
#pragma unroll
  for (int d = 0; d < DF; ++d) {
    float xv = x[i * DF + d];
    a += xv * mlp_w[d];
    b += xv * mlp_w[DF + d];
  }
  s_node[i] = a;
  s_nbr[i]  = b;
}

// ---------------------------------------------------------------------------
// 2) per-edge score: e_val = edge_attr @ w_edge
// ---------------------------------------------------------------------------
__global__ void k_edge_val(const float* __restrict__ edge_attr, const float* __restrict__ mlp_w,
                           float* __restrict__ e_val, int E) {
  int e = blockIdx.x * blockDim.x + threadIdx.x;
  if (e >= E) return;
  float a = 0.f;
#pragma unroll
  for (int d = 0; d < DF; ++d) a += edge_attr[e * DF + d] * mlp_w[2 * DF + d];
  e_val[e] = a;
}

// ---------------------------------------------------------------------------
// 3) dense score init to NEG (64 MB, fits MI455X 192MB L2) — float4 stores
// ---------------------------------------------------------------------------
__global__ void k_init_S(float4* __restrict__ S, size_t n4) {
  size_t i = (size_t)blockIdx.x * blockDim.x + threadIdx.x;
  size_t stride = (size_t)gridDim.x * blockDim.x;
  float4 v = make_float4(NEGV, NEGV, NEGV, NEGV);
  for (; i < n4; i += stride) S[i] = v;
}

// generic float fill
__global__ void k_fill(float* __restrict__ p, float v, int n) {
  int i = blockIdx.x * blockDim.x + threadIdx.x;
  if (i < n) p[i] = v;
}

// ---------------------------------------------------------------------------
// 4) scatter neighbor scores: S[r,c] = s_node[r] + s_nbr[c] + e_val[e] + b
//    (plain store: duplicate (r,c) edges -> one writer wins, same as .set)
// ---------------------------------------------------------------------------
__global__ void k_scatter(const long long* __restrict__ ei, const float* __restrict__ e_val,
                          const float* __restrict__ s_node, const float* __restrict__ s_nbr,
                          const float* __restrict__ mlp_b, float* __restrict__ S, int E, int N) {
  int e = blockIdx.x * blockDim.x + threadIdx.x;
  if (e >= E) return;
  long long r = ei[e], c = ei[E + e];
  S[(size_t)r * N + c] = s_node[r] + s_nbr[c] + e_val[e] + mlp_b[0];
}

// ---------------------------------------------------------------------------
// 5) fused: double softmax + top-16 per row. One block per row, 16 vals/thread
//    kept in registers; LDS tree reductions; iterative argmax extraction with
//    tie-break on smaller column (matches jax.lax.top_k).
// ---------------------------------------------------------------------------
__global__ void __launch_bounds__(ROWB) k_row(const float* __restrict__ S,
                                              float* __restrict__ kth,
                                              int* __restrict__ topk, int N) {
  __shared__ float rf[ROWB];
  __shared__ int   rc[ROWB];
  const int i = blockIdx.x;
  const int t = threadIdx.x;
  const int PT = 16;  // N / ROWB (N == 4096)

  float z[PT];
  int   col[PT];
  const float* row = S + (size_t)i * N;
#pragma unroll
  for (int k = 0; k < PT; ++k) { col[k] = k * ROWB + t; z[k] = row[col[k]]; }

  // neighbor flags before values get overwritten
  unsigned nbm = 0;
#pragma unroll
  for (int k = 0; k < PT; ++k) if (z[k] > -1.0e8f) nbm |= 1u << k;

  // ---- softmax #1 ----
  float m = -3.4e38f;
#pragma unroll
  for (int k = 0; k < PT; ++k) m = fmaxf(m, z[k]);
  rf[t] = m; __syncthreads();
  for (int s = ROWB / 2; s > 0; s >>= 1) { if (t < s) rf[t] = fmaxf(rf[t], rf[t + s]); __syncthreads(); }
  m = rf[0]; __syncthreads();

  float sum = 0.f;
#pragma unroll
  for (int k = 0; k < PT; ++k) { z[k] = expf(z[k] - m); sum += z[k]; }
  rf[t] = sum; __syncthreads();
  for (int s = ROWB / 2; s > 0; s >>= 1) { if (t < s) rf[t] += rf[t + s]; __syncthreads(); }
  sum = rf[0]; __syncthreads();
  float inv = 1.0f / sum;
#pragma unroll
  for (int k = 0; k < PT; ++k) z[k] *= inv;   // z1 (non-neighbors already exp(-1e9)=0)

  // ---- softmax #2 over (neighbor ? z1 : NEG) ----
  float m2 = -3.4e38f;
#pragma unroll
  for (int k = 0; k < PT; ++k) m2 = fmaxf(m2, ((nbm >> k) & 1u) ? z[k] : NEGV);
  rf[t] = m2; __syncthreads();
  for (int s = ROWB / 2; s > 0; s >>= 1) { if (t < s) rf[t] = fmaxf(rf[t], rf[t + s]); __syncthreads(); }
  m2 = rf[0]; __syncthreads();

  float sum2 = 0.f;
#pragma unroll
  for (int k = 0; k < PT; ++k) {
    z[k] = ((nbm >> k) & 1u) ? expf(z[k] - m2) : 0.0f;
    sum2 += z[k];
  }
  rf[t] = sum2; __syncthreads();
  for (int s = ROWB / 2; s > 0; s >>= 1) { if (t < s) rf[t] += rf[t + s]; __syncthreads(); }
  sum2 = rf[0]; __syncthreads();
  float inv2 = 1.0f / sum2;
#pragma unroll
  for (int k = 0; k < PT; ++k) z[k] *= inv2;  // z2; non-neighbors exactly 0

  // ---- iterative top-16 extraction ----
  unsigned used = 0;
  for (int r = 0; r < KSEL; ++r) {
    float bv = -1.0f; int bc = 0x7fffffff; int bk = -1;
#pragma unroll
    for (int k = 0; k < PT; ++k) {
      if (!((used >> k) & 1u)) {
        float v = z[k];
        if (v > bv || (v == bv && col[k] < bc)) { bv = v; bc = col[k]; bk = k; }
      }
    }
    rf[t] = bv; rc[t] = bc; __syncthreads();
    for (int s = ROWB / 2; s > 0; s >>= 1) {
      if (t < s) {
        if (rf[t + s] > rf[t] || (rf[t + s] == rf[t] && rc[t + s] < rc[t])) {
          rf[t] = rf[t + s]; rc[t] = rc[t + s];
        }
      }
      __syncthreads();
    }
    float wv = rf[0]; int wc = rc[0]; __syncthreads();
    if (wc == bc && bk >= 0) used |= 1u << bk;   // winner thread retires its slot
    if (t == 0) {
      topk[i * KSEL + r] = wc;
      if (r == KSEL - 1) kth[i] = wv;
    }
  }
}

// ---------------------------------------------------------------------------
// 6) per-edge keep mask + degree accumulation (deg pre-filled with 1.0)
// ---------------------------------------------------------------------------
__global__ void k_keep(const long long* __restrict__ ei, const int* __restrict__ node_mask,
                       const float* __restrict__ kth, const int* __restrict__ topk,
                       int* __restrict__ keep, float* __restrict__ deg, int E) {
  int e = blockIdx.x * blockDim.x + threadIdx.x;
  if (e >= E) return;
  long long r = ei[e], c = ei[E + e];
  int kk = 0;
  if (node_mask[r] != 0 && kth[r] != 0.0f) {
    const int* tr = topk + r * KSEL;
#pragma unroll
    for (int j = 0; j < KSEL; ++j) kk |= (tr[j] == (int)c);
  }
  keep[e] = kk;
  if (kk) atomicAdd(&deg[c], 1.0f);
}

__global__ void k_dinv(const float* __restrict__ deg, float* __restrict__ dinv, int N) {
  int i = blockIdx.x * blockDim.x + threadIdx.x;
  if (i < N) dinv[i] = rsqrtf(deg[i]);
}

// ---------------------------------------------------------------------------
// 7) f32 GEMM via V_WMMA_F32_16X16X4_F32: one wave per 16x16 output tile.
//    A: M x Kd row-major, B: Kd x Nc row-major. VGPR striping per ISA 7.12.2:
//    A lane = (M % 16), half = lane>>4 selects K pair; C/D: vgpr v -> row v+8*half.
// ---------------------------------------------------------------------------
__global__ void k_gemm_wmma(const float* __restrict__ A, const float* __restrict__ B,
                            const float* __restrict__ bias, float* __restrict__ C,
                            int M, int Kd, int Nc, int tilesN, int relu) {
  int gw   = (int)((blockIdx.x * blockDim.x + threadIdx.x) >> 5);
  int lane = threadIdx.x & 31;
  int tilesM = M >> 4;
  if (gw >= tilesM * tilesN) return;
  int tm = gw / tilesN, tn = gw % tilesN;

  int mrow = lane & 15;
  int half = lane >> 4;

  v8f acc = {};
  const float* Arow = A + (size_t)(tm * 16 + mrow) * Kd;
  int n = tn * 16 + mrow;  // B/C column index for this lane

  for (int k0 = 0; k0 < Kd; k0 += 4) {
    v2f a, b;
    a[0] = Arow[k0 + 2 * half + 0];
    a[1] = Arow[k0 + 2 * half + 1];
    b[0] = B[(size_t)(k0 + 2 * half + 0) * Nc + n];
    b[1] = B[(size_t)(k0 + 2 * half + 1) * Nc + n];
    acc = __builtin_amdgcn_wmma_f32_16x16x4_f32(false, a, false, b,
                                                (short)0, acc, false, false);
  }

#pragma unroll
  for (int v = 0; v < 8; ++v) {
    int row = tm * 16 + v + 8 * half;
    float r = acc[v] + (bias ? bias[n] : 0.0f);
    if (relu) r = fmaxf(r, 0.0f);
    C[(size_t)row * Nc + n] = r;
  }
}

// ---------------------------------------------------------------------------
// 8) sparse aggregation: agg[c,:] += dinv[r]*dinv[c] * hW[r,:] over kept edges
// ---------------------------------------------------------------------------
__global__ void k_agg(const long long* __restrict__ ei, const int* __restrict__ keep,
                      const float* __restrict__ dinv, const float* __restrict__ hW,
                      float* __restrict__ agg, int E) {
  int e = blockIdx.x * blockDim.x + threadIdx.x;
  if (e >= E || !keep[e]) return;
  long long r = ei[e], c = ei[E + e];
  float coef = dinv[r] * dinv[c];
#pragma unroll
  for (int h = 0; h < HF; ++h) atomicAdd(&agg[(size_t)c * HF + h], coef * hW[(size_t)r * HF + h]);
}

// 9) combine: out = relu(agg + dinv^2 * hW + b)
__global__ void k_combine(const float* __restrict__ agg, const float* __restrict__ hW,
                          const float* __restrict__ dinv, const float* __restrict__ bias,
                          float* __restrict__ out, int N) {
  int idx = blockIdx.x * blockDim.x + threadIdx.x;
  if (idx >= N * HF) return;
  int i = idx / HF, h = idx % HF;
  float di = dinv[i];
  float v = agg[idx] + di * di * hW[idx] + bias[h];
  out[idx] = fmaxf(v, 0.0f);
}

// ---------------------------------------------------------------------------
extern "C" void kernel_launch(void* const* d_in, const int* in_sizes, int n_in,
                              void* d_out, int out_size, void* d_ws, size_t ws_size,
                              hipStream_t stream) {
  // input order: num_nodes, edge_index, edge_attr, x, node_mask, mlp_w, mlp_b,
  //              w1, b1, w2, b2, fc_w, fc_b
  const long long* edge_index = (const long long*)d_in[1];
  const float* edge_attr = (const float*)d_in[2];
  const float* x         = (const float*)d_in[3];
  const int*   node_mask = (const int*)d_in[4];
  const float* mlp_w     = (const float*)d_in[5];
  const float* mlp_b     = (const float*)d_in[6];
  const float* w1        = (const float*)d_in[7];
  const float* b1        = (const float*)d_in[8];
  const float* w2        = (const float*)d_in[9];
  const float* b2        = (const float*)d_in[10];
  const float* fc_w      = (const float*)d_in[11];
  const float* fc_b      = (const float*)d_in[12];
  float* out = (float*)d_out;

  const int N = in_sizes[3] / DF;     // 4096
  const int E = in_sizes[1] / 2;      // 131072

  // workspace carve-up (floats)
  float* S      = (float*)d_ws;                     // N*N
  float* e_val  = S + (size_t)N * N;                // E
  float* s_node = e_val + E;                        // N
  float* s_nbr  = s_node + N;                       // N
  float* kth    = s_nbr + N;                        // N
  int*   topk   = (int*)(kth + N);                  // N*KSEL
  int*   keep   = topk + (size_t)N * KSEL;          // E
  float* deg    = (float*)(keep + E);               // N
  float* dinv   = deg + N;                          // N
  float* hW1    = dinv + N;                         // N*HF
  float* h1     = hW1 + (size_t)N * HF;             // N*HF
  float* hW2    = h1 + (size_t)N * HF;              // N*HF
  float* h2     = hW2 + (size_t)N * HF;             // N*HF
  float* agg    = h2 + (size_t)N * HF;              // N*HF

  const int TB = 256;
  int gN  = (N + TB - 1) / TB;
  int gE  = (E + TB - 1) / TB;
  int gNH = (N * HF + TB - 1) / TB;

  // ---- phase 1: scores ----
  k_node_scores<<<gN, TB, 0, stream>>>(x, mlp_w, s_node, s_nbr, N);
  k_edge_val<<<gE, TB, 0, stream>>>(edge_attr, mlp_w, e_val, E);
  k_init_S<<<2048, TB, 0, stream>>>((float4*)S, (size_t)N * N / 4);
  k_scatter<<<gE, TB, 0, stream>>>(edge_index, e_val, s_node, s_nbr, mlp_b, S, E, N);

  // ---- phase 2: fused double-softmax + top-16 ----
  k_row<<<N, ROWB, 0, stream>>>(S, kth, topk, N);

  // ---- phase 3: edge keep + degree ----
  k_fill<<<gN, TB, 0, stream>>>(deg, 1.0f, N);
  k_keep<<<gE, TB, 0, stream>>>(edge_index, node_mask, kth, topk, keep, deg, E);
  k_dinv<<<gN, TB, 0, stream>>>(deg, dinv, N);

  // ---- GCN layer 1: hW1 = x @ w1 (WMMA), aggregate, combine+relu ----
  {
    int tilesM = N / 16, tilesN = 1;
    int waves = tilesM * tilesN;
    int grid = (waves * 32 + TB - 1) / TB;
    k_gemm_wmma<<<grid, TB, 0, stream>>>(x, w1, nullptr, hW1, N, DF, HF, tilesN, 0);
  }
  k_fill<<<gNH, TB, 0, stream>>>(agg, 0.0f, N * HF);
  k_agg<<<gE, TB, 0, stream>>>(edge_index, keep, dinv, hW1, agg, E);
  k_combine<<<gNH, TB, 0, stream>>>(agg, hW1, dinv, b1, h1, N);

  // ---- GCN layer 2: hW2 = h1 @ w2 (WMMA), aggregate, combine+relu ----
  {
    int tilesM = N / 16, tilesN = 1;
    int waves = tilesM * tilesN;
    int grid = (waves * 32 + TB - 1) / TB;
    k_gemm_wmma<<<grid, TB, 0, stream>>>(h1, w2, nullptr, hW2, N, HF, HF, tilesN, 0);
  }
  k_fill<<<gNH, TB, 0, stream>>>(agg, 0.0f, N * HF);
  k_agg<<<gE, TB, 0, stream>>>(edge_index, keep, dinv, hW2, agg, E);
  k_combine<<<gNH, TB, 0, stream>>>(agg, hW2, dinv, b2, h2, N);

  // ---- final: out = h2 @ fc_w + fc_b (WMMA, 7 column tiles) ----
  {
    int tilesM = N / 16, tilesN = OUTF / 16;   // 256 x 7
    int waves = tilesM * tilesN;
    int grid = (waves * 32 + TB - 1) / TB;
    k_gemm_wmma<<<grid, TB, 0, stream>>>(h2, fc_w, fc_b, out, N, HF, OUTF, tilesN, 0);
  }
}